// Set2Set_18880676233593
// MI455X (gfx1250) — compile-verified
//
#include <hip/hip_runtime.h>
#include <hip/hip_bf16.h>
#include <math.h>

// Problem constants (fixed by the reference)
#define HDIM 256      // hidden size H
#define NGRAPH 2048   // B (num_graphs; harness passes it as a device scalar, but it is fixed)
#define MITER 6

typedef __attribute__((ext_vector_type(2))) float v2f;
typedef __attribute__((ext_vector_type(8))) float v8f;

// ---------------------------------------------------------------------------
// Segment offsets from sorted graph_id: off[g] = first index with id >= g.
// ---------------------------------------------------------------------------
__global__ void offsets_kernel(const int* __restrict__ gid, int* __restrict__ off, int N) {
    int i = blockIdx.x * blockDim.x + threadIdx.x;
    if (i >= N) return;
    int cur  = gid[i];
    int prev = (i == 0) ? -1 : gid[i - 1];
    for (int g = prev + 1; g <= cur; ++g) off[g] = i;
    if (i == N - 1) {
        for (int g = cur + 1; g <= NGRAPH; ++g) off[g] = N;
    }
}

__global__ void zero_kernel(float* __restrict__ p, int n) {
    int i = blockIdx.x * blockDim.x + threadIdx.x;
    if (i < n) p[i] = 0.0f;
}

// ---------------------------------------------------------------------------
// Fused attention + softmax + weighted readout, one block (8 wave32) per graph.
// Writes q_star[g] = [ q[g] | r[g] ]  (row of 2H floats).
// ---------------------------------------------------------------------------
__global__ __launch_bounds__(256)
void attention_kernel(const float* __restrict__ nodes,
                      const int*   __restrict__ offsets,
                      const float* __restrict__ q,
                      float*       __restrict__ e_ws,
                      float*       __restrict__ q_star) {
    const int g    = blockIdx.x;
    const int tid  = threadIdx.x;        // 0..255
    const int wave = tid >> 5;
    const int lane = tid & 31;

    __shared__ float q_s[HDIM];
    __shared__ float wmax[8];

    const int start = offsets[g];
    const int end   = offsets[g + 1];

    q_s[tid] = q[(size_t)g * HDIM + tid];
    __syncthreads();

    // Pass 1: e_v = <nodes_v, q_g>, track running max. One wave per node.
    float m = -INFINITY;
    for (int n = start + wave; n < end; n += 8) {
        const float* np = nodes + (size_t)n * HDIM;
        float s = 0.0f;
        #pragma unroll
        for (int d = 0; d < HDIM; d += 32) s += np[d + lane] * q_s[d + lane];
        #pragma unroll
        for (int off = 16; off > 0; off >>= 1) s += __shfl_xor(s, off, 32);
        if (lane == 0) e_ws[n] = s;      // s is identical in all lanes now
        m = fmaxf(m, s);
    }
    if (lane == 0) wmax[wave] = m;
    __threadfence();                      // make e_ws writes visible block-wide
    __syncthreads();

    float emax = -INFINITY;
    #pragma unroll
    for (int w = 0; w < 8; ++w) emax = fmaxf(emax, wmax[w]);

    // Pass 2: r_d = sum_n exp(e_n - emax) * nodes[n][d] / denom. Thread owns dim d=tid.
    float racc  = 0.0f;
    float denom = 0.0f;
    for (int n = start; n < end; ++n) {
        float ex = expf(e_ws[n] - emax);  // broadcast load, redundant exp (cheap)
        denom += ex;
        racc  += ex * nodes[(size_t)n * HDIM + tid];
    }
    float rv = (end > start) ? (racc / denom) : 0.0f;

    q_star[(size_t)g * (2 * HDIM) + tid]        = q_s[tid];
    q_star[(size_t)g * (2 * HDIM) + HDIM + tid] = rv;
}

// ---------------------------------------------------------------------------
// LSTM gate GEMM via V_WMMA_F32_16X16X4_F32:
//   C[M x Ncols] = A1[M x K1] @ W1^T + A2[M x K2] @ W2^T
// W row-major [Ncols x K], so B-matrix element B[k][n] = W[n][k].
// One wave computes one 16x16 tile; 8 waves per block.
// f32 A 16x4 layout: lanes 0-15 hold {K=0,1}, lanes 16-31 hold {K=2,3}; B mirrors it.
// ---------------------------------------------------------------------------
__device__ inline void wmma_k_loop(v8f& acc, const float* __restrict__ arow,
                                   const float* __restrict__ wrow, int K, int half) {
    for (int k0 = 0; k0 < K; k0 += 4) {
        const int k = k0 + 2 * half;
        v2f a; a.x = arow[k]; a.y = arow[k + 1];
        v2f b; b.x = wrow[k]; b.y = wrow[k + 1];
        acc = __builtin_amdgcn_wmma_f32_16x16x4_f32(
            /*neg_a=*/false, a, /*neg_b=*/false, b,
            /*c_mod=*/(short)0, acc, /*reuse_a=*/false, /*reuse_b=*/false);
    }
}

__global__ __launch_bounds__(256)
void lstm_gates_wmma(const float* __restrict__ A1, int K1, const float* __restrict__ W1,
                     const float* __restrict__ A2, int K2, const float* __restrict__ W2,
                     float* __restrict__ C, int M, int Ncols) {
    const int tid  = threadIdx.x;
    const int wave = tid >> 5;
    const int lane = tid & 31;
    const int tile = blockIdx.x * 8 + wave;
    const int ntN  = Ncols >> 4;
    const int rowBase = (tile / ntN) << 4;
    const int colBase = (tile % ntN) << 4;
    if (rowBase >= M) return;            // wave-uniform; grid is sized exactly anyway

    const int half = lane >> 4;          // 0: K pair {0,1}; 1: K pair {2,3}
    const int l    = lane & 15;

    v8f acc = {};
    wmma_k_loop(acc, A1 + (size_t)(rowBase + l) * K1,
                     W1 + (size_t)(colBase + l) * K1, K1, half);
    wmma_k_loop(acc, A2 + (size_t)(rowBase + l) * K2,
                     W2 + (size_t)(colBase + l) * K2, K2, half);

    // D layout: VGPR r -> row (rowBase + r + 8*half), col (colBase + l)
    float* crow = C + (size_t)(rowBase + half * 8) * Ncols + colBase + l;
    #pragma unroll
    for (int r = 0; r < 8; ++r) crow[(size_t)r * Ncols] = acc[r];
}

// ---------------------------------------------------------------------------
// LSTM elementwise: gates (i,f,g,o order) + biases -> new h, c.
// ---------------------------------------------------------------------------
__global__ void lstm_update_kernel(const float* __restrict__ gates,
                                   const float* __restrict__ bih,
                                   const float* __restrict__ bhh,
                                   float* __restrict__ h, float* __restrict__ c) {
    int idx = blockIdx.x * blockDim.x + threadIdx.x;   // over B*H
    if (idx >= NGRAPH * HDIM) return;
    int b = idx >> 8;        // /HDIM
    int j = idx & (HDIM - 1);
    const float* grow = gates + (size_t)b * (4 * HDIM);
    float xi = grow[j]            + bih[j]            + bhh[j];
    float xf = grow[HDIM + j]     + bih[HDIM + j]     + bhh[HDIM + j];
    float xg = grow[2 * HDIM + j] + bih[2 * HDIM + j] + bhh[2 * HDIM + j];
    float xo = grow[3 * HDIM + j] + bih[3 * HDIM + j] + bhh[3 * HDIM + j];
    float ig = 1.0f / (1.0f + expf(-xi));
    float fg = 1.0f / (1.0f + expf(-xf));
    float gg = tanhf(xg);
    float og = 1.0f / (1.0f + expf(-xo));
    float cn = fg * c[idx] + ig * gg;
    h[idx] = og * tanhf(cn);
    c[idx] = cn;
}

// ---------------------------------------------------------------------------
// Launch
// ---------------------------------------------------------------------------
static inline size_t align256(size_t x) { return (x + 255) & ~(size_t)255; }

extern "C" void kernel_launch(void* const* d_in, const int* in_sizes, int n_in,
                              void* d_out, int out_size, void* d_ws, size_t ws_size,
                              hipStream_t stream) {
    const float* nodes    = (const float*)d_in[0];
    const int*   graph_id = (const int*)d_in[1];   // harness delivers integers as int32
    // d_in[2] = num_graphs scalar on device; fixed at 2048 for this problem
    const float* Wih0 = (const float*)d_in[3];
    const float* Whh0 = (const float*)d_in[4];
    const float* bih0 = (const float*)d_in[5];
    const float* bhh0 = (const float*)d_in[6];
    const float* Wih1 = (const float*)d_in[7];
    const float* Whh1 = (const float*)d_in[8];
    const float* bih1 = (const float*)d_in[9];
    const float* bhh1 = (const float*)d_in[10];

    const int N = in_sizes[0] / HDIM;    // 200000
    float* out = (float*)d_out;

    // Workspace carve-up
    char* ws = (char*)d_ws;
    int*   offsets = (int*)ws;                    ws += align256((NGRAPH + 1) * sizeof(int));
    float* e_ws    = (float*)ws;                  ws += align256((size_t)N * sizeof(float));
    float* state   = (float*)ws;                  ws += align256((size_t)5 * NGRAPH * HDIM * sizeof(float));
    float* q0 = state;                            // zero-initialized q for iteration 0
    float* h0 = state + 1 * NGRAPH * HDIM;
    float* c0 = state + 2 * NGRAPH * HDIM;
    float* h1 = state + 3 * NGRAPH * HDIM;
    float* c1 = state + 4 * NGRAPH * HDIM;
    float* q_star = (float*)ws;                   ws += align256((size_t)NGRAPH * 2 * HDIM * sizeof(float));
    float* gates  = (float*)ws;                   ws += align256((size_t)NGRAPH * 4 * HDIM * sizeof(float));
    (void)ws_size; (void)n_in; (void)out_size;

    // Segment offsets + zero recurrent state (fresh every call -> deterministic)
    offsets_kernel<<<(N + 255) / 256, 256, 0, stream>>>(graph_id, offsets, N);
    {
        int nz = 5 * NGRAPH * HDIM;
        zero_kernel<<<(nz + 255) / 256, 256, 0, stream>>>(state, nz);
    }

    const int Mrows = NGRAPH;            // 2048
    const int Ncols = 4 * HDIM;          // 1024
    const int tiles = (Mrows / 16) * (Ncols / 16);   // 8192
    const int gemm_blocks = tiles / 8;               // 1024

    for (int it = 0; it < MITER; ++it) {
        const float* qptr = (it == 0) ? q0 : h1;
        float* qs_out = (it == MITER - 1) ? out : q_star;
        attention_kernel<<<NGRAPH, 256, 0, stream>>>(nodes, offsets, qptr, e_ws, qs_out);
        if (it < MITER - 1) {
            // Layer 0: gates = q_star @ Wih0^T + h0 @ Whh0^T
            lstm_gates_wmma<<<gemm_blocks, 256, 0, stream>>>(
                q_star, 2 * HDIM, Wih0, h0, HDIM, Whh0, gates, Mrows, Ncols);
            lstm_update_kernel<<<(NGRAPH * HDIM + 255) / 256, 256, 0, stream>>>(
                gates, bih0, bhh0, h0, c0);
            // Layer 1: gates = h0 @ Wih1^T + h1 @ Whh1^T
            lstm_gates_wmma<<<gemm_blocks, 256, 0, stream>>>(
                h0, HDIM, Wih1, h1, HDIM, Whh1, gates, Mrows, Ncols);
            lstm_update_kernel<<<(NGRAPH * HDIM + 255) / 256, 256, 0, stream>>>(
                gates, bih1, bhh1, h1, c1);
        }
    }
}